// STFT_36704790511975
// MI455X (gfx1250) — compile-verified
//
#include <hip/hip_runtime.h>
#include <math.h>

#define FILT   1024
#define HOPS   512
#define CUT    513
#define NCH    1026
#define MPAD   1040          // 65 tiles of 16
#define BATCH  8
#define NSAMP  2097152
#define NFRAME 4097          // valid frames
#define TPAD   4160          // padded frame count (multiple of 64)
#define PI_F   3.14159265358979323846f

typedef float v2f __attribute__((ext_vector_type(2)));
typedef float v8f __attribute__((ext_vector_type(8)));

static __device__ __forceinline__ v8f wmma_f32(v2f a, v2f b, v8f c) {
    // V_WMMA_F32_16X16X4_F32 : D = A(16x4) * B(4x16) + C(16x16), all f32
    return __builtin_amdgcn_wmma_f32_16x16x4_f32(false, a, false, b, (short)0, c, false, false);
}

// ---------------------------------------------------------------------------
// Kernel 1: synthesize bases analytically, stored K-pair-interleaved so that
// WMMA A-fragments are single coalesced b64 loads:
//   fbP [(k>>1)][c][k&1]  = fb[c][k]          (forward basis, A of GEMM1)
//   invP[(c>>1)][j][c&1]  = inv[c][j]         (pinv basis,   A of GEMM2)
// ---------------------------------------------------------------------------
__global__ void build_bases(float* __restrict__ fbP, float* __restrict__ invP) {
    int idx = blockIdx.x * blockDim.x + threadIdx.x;
    if (idx >= MPAD * FILT) return;
    int p = idx & (FILT - 1);   // k for fb, j for inv  (0..1023)
    int q = idx >> 10;          // channel c            (0..1039)
    const float C = 2.0f * PI_F / (float)FILT;
    float s  = sinf((float)p * (PI_F / (float)FILT));
    float wk = s * s;           // Hann window, relative-accurate near zeros

    // ---- forward basis: fb[c][k] = w_k * {cos, -sin}(2*pi*c*k/1024) ----
    float fbv = 0.0f;
    if (q < CUT) {
        int r = (q * p) & (FILT - 1);
        fbv = wk * cosf((float)r * C);
    } else if (q < NCH) {
        int r = ((q - CUT) * p) & (FILT - 1);
        fbv = -wk * sinf((float)r * C);
    }
    fbP[(size_t)(p >> 1) * (2 * MPAD) + q * 2 + (p & 1)] = fbv;

    // ---- analytic pinv: inv[c][j] = 0.5*pinv(fb)[j][c]
    // fb^T fb = 512*diag(w^2) + p_e p_e^T + p_o p_o^T (disjoint parity support)
    float v = 0.0f;
    if (p > 0 && q < NCH) {
        float inv512w = 1.0f / (512.0f * wk);
        if (q < CUT) {
            int r = (q * p) & (FILT - 1);
            float base = cosf((float)r * C) * inv512w;
            int par = p & 1;
            float ssum, denomN;
            if (par) { ssum = (q == 0) ? 512.0f : ((q == 512) ? -512.0f : 0.0f); denomN = 1024.0f; }
            else     { ssum = (q == 0 || q == 512) ? 511.0f : -1.0f;             denomN = 1023.0f; }
            v = base - ssum * inv512w / denomN;
        } else {
            int r = ((q - CUT) * p) & (FILT - 1);
            v = -sinf((float)r * C) * inv512w;   // sine correction sums vanish
        }
    }
    invP[(size_t)(q >> 1) * (2 * FILT) + p * 2 + (q & 1)] = 0.5f * v;
}

// ---------------------------------------------------------------------------
// Kernel 2: forward STFT GEMM.  spec[b][c][t] = sum_k fb[c][k] * X[k][t]
// 8 waves x (16M x 64N) per block. Shared 65*512-sample window in LDS with
// +2/512-block swizzle; k-loop split at 512 so all LDS addressing is linear.
// ---------------------------------------------------------------------------
#define TG1       64                       // frames per block (4 N-tiles/wave)
#define LDSEG     65                       // ((TG1-1)*HOP + FILT)/HOP
#define LDSN      (LDSEG * HOPS)           // 33280 payload floats
#define LDSTOT    (LDSN + 2 * LDSEG)       // 33410 floats = 133640 B

__global__ __launch_bounds__(256)
void stft_fwd(const float* __restrict__ wav, const float* __restrict__ fbP,
              float* __restrict__ spec) {
    __shared__ float xs[LDSTOT];
    const int b  = blockIdx.y;
    const int tg = blockIdx.x;                      // frames [64*tg, 64*tg+64)
    const long base = (long)tg * TG1 * HOPS - HOPS;
    const float* wavb = wav + (size_t)b * NSAMP;

    for (int i = threadIdx.x; i < LDSN; i += 256) {
        long sIdx = base + i;
        float v = (sIdx >= 0 && sIdx < NSAMP) ? wavb[sIdx] : 0.0f;
        xs[i + 2 * (i >> 9)] = v;
    }
    __syncthreads();

    const int lane  = threadIdx.x & 31;
    const int wv    = threadIdx.x >> 5;
    const int mTile = blockIdx.z * 8 + wv;
    if (mTile >= MPAD / 16) return;                 // wave-uniform, after barrier
    const int c0    = mTile * 16;
    const int mrow  = lane & 15;                    // A: M row / B,D: N col
    const int khalf = (lane >> 4) << 1;             // lanes 16-31 hold K+2,K+3

    v8f acc[4] = {};
    // A: one coalesced b64 per k-step from pair-interleaved basis
    const float* aP = fbP + (size_t)(khalf >> 1) * (2 * MPAD) + 2 * (c0 + mrow);
    // B: swizzled LDS pointers, linear within each 512-half of k
    const float* p0 = xs + (mrow     ) * HOPS + khalf + 2 * (mrow     );
    const float* p1 = xs + (mrow + 16) * HOPS + khalf + 2 * (mrow + 16);
    const float* p2 = xs + (mrow + 32) * HOPS + khalf + 2 * (mrow + 32);
    const float* p3 = xs + (mrow + 48) * HOPS + khalf + 2 * (mrow + 48);

    #pragma unroll 4
    for (int k0 = 0; k0 < 512; k0 += 4) {
        v2f a  = *(const v2f*)(aP + (size_t)k0 * MPAD);
        v2f b0 = *(const v2f*)(p0 + k0);
        v2f b1 = *(const v2f*)(p1 + k0);
        v2f b2 = *(const v2f*)(p2 + k0);
        v2f b3 = *(const v2f*)(p3 + k0);
        acc[0] = wmma_f32(a, b0, acc[0]);
        acc[1] = wmma_f32(a, b1, acc[1]);
        acc[2] = wmma_f32(a, b2, acc[2]);
        acc[3] = wmma_f32(a, b3, acc[3]);
    }
    #pragma unroll 4
    for (int k0 = 512; k0 < 1024; k0 += 4) {        // swizzle crossed: +2
        v2f a  = *(const v2f*)(aP + (size_t)k0 * MPAD);
        v2f b0 = *(const v2f*)(p0 + k0 + 2);
        v2f b1 = *(const v2f*)(p1 + k0 + 2);
        v2f b2 = *(const v2f*)(p2 + k0 + 2);
        v2f b3 = *(const v2f*)(p3 + k0 + 2);
        acc[0] = wmma_f32(a, b0, acc[0]);
        acc[1] = wmma_f32(a, b1, acc[1]);
        acc[2] = wmma_f32(a, b2, acc[2]);
        acc[3] = wmma_f32(a, b3, acc[3]);
    }

    // D layout: VGPR r -> rows (r, r+8) split across lane halves, col = lane&15
    float* sp = spec + (size_t)b * MPAD * TPAD;
    const int cw = c0 + (lane >> 4) * 8;
    const int tb = tg * TG1 + (lane & 15);
    #pragma unroll
    for (int n = 0; n < 4; ++n)
        #pragma unroll
        for (int r = 0; r < 8; ++r)
            sp[(size_t)(cw + r) * TPAD + tb + 16 * n] = acc[n][r];
}

// ---------------------------------------------------------------------------
// Kernel 3: inverse GEMM + overlap-add.
// Y[j][t] = sum_c inv[c][j]*spec[c][t];  out[512*t-512+j] += Y[j][t].
// B tile (16 c-rows x 64 parity-cols) staged in LDS once per block (shared by
// all 8 waves). Two launches: even t stores, odd t accumulates -> race-free
// and deterministic without atomics.
// ---------------------------------------------------------------------------
#define TG2 64                               // parity-columns per block

__global__ __launch_bounds__(256)
void stft_inv(const float* __restrict__ invP, const float* __restrict__ spec,
              float* __restrict__ out, int parity) {
    __shared__ float bs[TG2 * 18];           // stride-18 pad: conflict-free b64
    const int b     = blockIdx.y;
    const int lane  = threadIdx.x & 31;
    const int wv    = threadIdx.x >> 5;
    const int j0    = (blockIdx.z * 8 + wv) * 16;
    const int mrow  = lane & 15;
    const int khalf = (lane >> 4) << 1;
    const int tq0   = blockIdx.x * TG2;      // base parity-column index

    const float* spb   = spec + (size_t)b * MPAD * TPAD;
    const float* aBase = invP + (size_t)(khalf >> 1) * (2 * FILT) + 2 * (j0 + mrow);

    v8f acc[4] = {};
    for (int cc = 0; cc < MPAD; cc += 16) {
        __syncthreads();                     // protect previous chunk's reads
        #pragma unroll
        for (int e = threadIdx.x; e < 16 * TG2; e += 256) {
            int kk  = e >> 6;                // 0..15
            int col = e & 63;
            int t   = parity + 2 * (tq0 + col);
            float v = (t < TPAD) ? spb[(size_t)(cc + kk) * TPAD + t] : 0.0f;
            bs[col * 18 + kk] = v;
        }
        __syncthreads();
        #pragma unroll
        for (int kc = 0; kc < 16; kc += 4) {
            v2f a  = *(const v2f*)(aBase + (size_t)(cc + kc) * FILT);
            int kl = kc + khalf;
            v2f b0 = *(const v2f*)&bs[(mrow     ) * 18 + kl];
            v2f b1 = *(const v2f*)&bs[(mrow + 16) * 18 + kl];
            v2f b2 = *(const v2f*)&bs[(mrow + 32) * 18 + kl];
            v2f b3 = *(const v2f*)&bs[(mrow + 48) * 18 + kl];
            acc[0] = wmma_f32(a, b0, acc[0]);
            acc[1] = wmma_f32(a, b1, acc[1]);
            acc[2] = wmma_f32(a, b2, acc[2]);
            acc[3] = wmma_f32(a, b3, acc[3]);
        }
    }

    float* outb = out + (size_t)b * NSAMP;
    const int jh = j0 + (lane >> 4) * 8;
    #pragma unroll
    for (int n = 0; n < 4; ++n) {
        int t = parity + 2 * (tq0 + 16 * n + (lane & 15));
        if (t < NFRAME) {
            #pragma unroll
            for (int r = 0; r < 8; ++r) {
                int m = t * HOPS - HOPS + jh + r;
                if (m >= 0 && m < NSAMP) {
                    if (parity) outb[m] += acc[n][r];
                    else        outb[m]  = acc[n][r];
                }
            }
        }
    }
}

// ---------------------------------------------------------------------------
extern "C" void kernel_launch(void* const* d_in, const int* in_sizes, int n_in,
                              void* d_out, int out_size, void* d_ws, size_t ws_size,
                              hipStream_t stream) {
    (void)in_sizes; (void)n_in; (void)out_size; (void)ws_size;
    const float* wav = (const float*)d_in[0];
    float* out = (float*)d_out;

    float* fbP  = (float*)d_ws;                         // FILT/2 pairs * 2*MPAD
    float* invP = fbP  + (size_t)MPAD * FILT;           // MPAD/2 pairs * 2*FILT
    float* spec = invP + (size_t)FILT * MPAD;           // BATCH * MPAD * TPAD

    int totalB = MPAD * FILT;
    build_bases<<<(totalB + 255) / 256, 256, 0, stream>>>(fbP, invP);

    dim3 g1(TPAD / TG1, BATCH, 9);                      // 65 x 8 x 9 (65 M-tiles)
    stft_fwd<<<g1, 256, 0, stream>>>(wav, fbP, spec);

    dim3 g2a(33, BATCH, 8);                             // even frames: t=0..4096
    stft_inv<<<g2a, 256, 0, stream>>>(invP, spec, out, 0);
    dim3 g2b(32, BATCH, 8);                             // odd frames:  t=1..4095
    stft_inv<<<g2b, 256, 0, stream>>>(invP, spec, out, 1);
}